// VanillaGNNLayer_38070590112569
// MI455X (gfx1250) — compile-verified
//
#include <hip/hip_runtime.h>
#include <hip/hip_bf16.h>

typedef __attribute__((ext_vector_type(2))) float v2f;
typedef __attribute__((ext_vector_type(8))) float v8f;

// -----------------------------------------------------------------------------
// Phase 1: h = x @ W^T  (full f32 via V_WMMA_F32_16X16X4_F32)
// One wave computes one 16x16 tile of h. Block = 256 threads = 8 waves =
// one 16-row stripe covering all 128 output features (8 col-tiles).
// Also writes the self-loop term into `out` (out starts as h).
// -----------------------------------------------------------------------------
__global__ void __launch_bounds__(256)
gnn_gemm_h_kernel(const float* __restrict__ x,   // [N,128]
                  const float* __restrict__ W,   // [128,128] (d_out, d_in)
                  float* __restrict__ h,         // [N,128]
                  float* __restrict__ out)       // [N,128]
{
    const int lane  = threadIdx.x & 31;
    const int wave  = threadIdx.x >> 5;      // 0..7  -> output-feature tile
    const int row0  = blockIdx.x * 16;       // 16-row stripe of x / h
    const int col0  = wave * 16;             // 16 output features

    const int l15   = lane & 15;
    const int khalf = (lane >> 4) << 1;      // 0 for lanes 0-15, 2 for 16-31

    // A fragment: lane reads x[row0 + l15, k + khalf + {0,1}]
    // B fragment: B = W^T, so lane reads W[col0 + l15, k + khalf + {0,1}]
    // (identical per-lane addressing pattern for A and B)
    const float* xrow = x + (size_t)(row0 + l15) * 128 + khalf;
    const float* wrow = W + (size_t)(col0 + l15) * 128 + khalf;

    v8f c = {};
#pragma unroll
    for (int k = 0; k < 128; k += 4) {
        v2f a = *(const v2f*)(xrow + k);
        v2f b = *(const v2f*)(wrow + k);
        // 8 args: (neg_a, A, neg_b, B, c_mod, C, reuse_a, reuse_b)
        c = __builtin_amdgcn_wmma_f32_16x16x4_f32(
                false, a, false, b, (short)0, c, false, false);
    }

    // 16x16 f32 D layout: lanes 0-15 -> M=r (VGPR r), N=lane;
    //                     lanes 16-31 -> M=r+8,      N=lane-16.
    const int half = lane >> 4;
    float* hp = h   + (size_t)(row0 + 8 * half) * 128 + col0 + l15;
    float* op = out + (size_t)(row0 + 8 * half) * 128 + col0 + l15;
#pragma unroll
    for (int r = 0; r < 8; ++r) {
        hp[(size_t)r * 128] = c[r];
        op[(size_t)r * 128] = c[r];
    }
}

// Scalar tail for row counts not divisible by 16 (unused for N=50000).
__global__ void gnn_gemm_tail_kernel(const float* __restrict__ x,
                                     const float* __restrict__ W,
                                     float* __restrict__ h,
                                     float* __restrict__ out,
                                     int rowStart, int N)
{
    int idx = blockIdx.x * blockDim.x + threadIdx.x;   // (row-rowStart)*128 + col
    int row = rowStart + (idx >> 7);
    int col = idx & 127;
    if (row >= N) return;
    const float* xr = x + (size_t)row * 128;
    const float* wr = W + (size_t)col * 128;
    float acc = 0.f;
#pragma unroll 8
    for (int k = 0; k < 128; ++k) acc = fmaf(xr[k], wr[k], acc);
    h[(size_t)row * 128 + col]   = acc;
    out[(size_t)row * 128 + col] = acc;
}

// -----------------------------------------------------------------------------
// Phase 2: out[dst] += h[src] for every edge. One wave per edge:
// 32 lanes x float4 = 128 channels. h/out are L2-resident (25.6 MB each of
// 192 MB), so gathers + no-return f32 atomics run at L2 bandwidth.
// -----------------------------------------------------------------------------
__global__ void __launch_bounds__(256)
gnn_scatter_kernel(const int* __restrict__ src,
                   const int* __restrict__ dst,
                   const float* __restrict__ h,
                   float* __restrict__ out,
                   int E)
{
    int gid  = blockIdx.x * blockDim.x + threadIdx.x;
    int e    = gid >> 5;
    if (e >= E) return;
    int lane = gid & 31;

    int s = src[e];
    int d = dst[e];

    float4 v = *((const float4*)(h + (size_t)s * 128) + lane);
    float* op = out + (size_t)d * 128 + lane * 4;
    atomicAdd(op + 0, v.x);
    atomicAdd(op + 1, v.y);
    atomicAdd(op + 2, v.z);
    atomicAdd(op + 3, v.w);
}

extern "C" void kernel_launch(void* const* d_in, const int* in_sizes, int n_in,
                              void* d_out, int out_size, void* d_ws, size_t ws_size,
                              hipStream_t stream)
{
    const float* x  = (const float*)d_in[0];   // [N,128]
    const int*   ei = (const int*)d_in[1];     // [2,E] flat: src then dst
    const float* W  = (const float*)d_in[2];   // [128,128]
    float* out = (float*)d_out;                // [N,128]
    float* h   = (float*)d_ws;                 // [N,128] scratch

    const int N = in_sizes[0] / 128;
    const int E = in_sizes[1] / 2;

    // Phase 1: GEMM (+ self-loop init of out)
    int fullTiles = N / 16;                    // 3125 for N=50000
    if (fullTiles > 0)
        gnn_gemm_h_kernel<<<fullTiles, 256, 0, stream>>>(x, W, h, out);
    int rem = N - fullTiles * 16;
    if (rem > 0) {
        int tailElems  = rem * 128;
        int tailBlocks = (tailElems + 255) / 256;
        gnn_gemm_tail_kernel<<<tailBlocks, 256, 0, stream>>>(x, W, h, out,
                                                             fullTiles * 16, N);
    }

    // Phase 2: edge scatter-add (one wave per edge)
    long long totalThreads = (long long)E * 32;
    int sblocks = (int)((totalThreads + 255) / 256);
    gnn_scatter_kernel<<<sblocks, 256, 0, stream>>>(ei, ei + E, h, out, E);
}